// PointNetAutoencoder_19181323944567
// MI455X (gfx1250) — compile-verified
//
#include <hip/hip_runtime.h>

// ---------------------------------------------------------------------------
// PointNet VN autoencoder forward, MI455X (gfx1250, wave32, WMMA).
// All channel-mixing GEMMs run through one strided batched WMMA kernel using
// V_WMMA_F32_16X16X4_F32 (fp32-exact; the network is memory bound at
// ~1 GB / 23.3 TB/s so low-precision matrix ops buy nothing).
// This revision makes the GEMM inner loop branch-free (clamped pointers +
// v_cndmask zeroing instead of exec-masked loads) and computes a 16x32 tile
// per wave (two accumulators sharing the A fragment).
// ---------------------------------------------------------------------------

#define EPSV     1e-6f
#define BN_EPSV  1e-5f
#define KNN      20
#define BATCH    16
#define NPTS     2048
#define MCOLS    (3 * NPTS)   // 6144 positions per batch for (C,3,N) tensors

typedef __attribute__((ext_vector_type(2))) float v2f;
typedef __attribute__((ext_vector_type(8))) float v8f;

// ===========================================================================
// Generic batched GEMM:  Y[b] (Cout x M) = W (Cout x Cin, ldW) * X[b]
// X element:  X[sXb*b + sXr*ci + sXc*m]
// Y element:  Y[sYb*b + o*M + m]   (+ rowBias[o]) (+ posBias[(b*Cout+o)*3 + m/Np])
// One wave32 computes a 16x32 f32 tile via 2x V_WMMA_F32_16X16X4_F32 per
// K-step of 4.  Fragment layouts (ISA 7.12.2):
//   A (16x4): lane m=l&15, kb=(l>>4)*2 -> {A[m][kb], A[m][kb+1]}
//   B (4x16): lane n=l&15, kb          -> {B[kb][n], B[kb+1][n]}
//   C/D:      vgpr i, lane l -> row i + (l>=16 ? 8 : 0), col l&15
// ===========================================================================
__global__ void gemm_wmma_kernel(const float* __restrict__ W, int ldW,
                                 const float* __restrict__ X,
                                 long long sXb, long long sXr, long long sXc,
                                 float* __restrict__ Y, long long sYb,
                                 const float* __restrict__ rowBias,
                                 const float* __restrict__ posBias,
                                 int Cout, int Cin, int M, int Np, int relu)
{
    const int lane = threadIdx.x & 31;
    const int wave = threadIdx.x >> 5;
    const int b    = blockIdx.z;
    const int row0 = blockIdx.y * 16;
    const int col0 = (blockIdx.x * 4 + wave) * 32;
    if (col0 >= M) return;                       // wave-uniform exit

    const int mloc = lane & 15;
    const int kb   = (lane >> 4) * 2;

    // Clamp indices once; loads below are unconditional, invalid lanes are
    // zeroed with selects (v_cndmask) -> no exec-mask branching in the loop.
    const int  r   = row0 + mloc;
    const bool rOk = (r < Cout);
    const float* Wr = W + (long long)(rOk ? r : 0) * ldW;

    const int  c0   = col0 + mloc;
    const int  c1   = col0 + 16 + mloc;
    const bool c0Ok = (c0 < M);
    const bool c1Ok = (c1 < M);
    const float* Xb  = X + (long long)b * sXb;
    const float* Xc0 = Xb + (long long)(c0Ok ? c0 : 0) * sXc;
    const float* Xc1 = Xb + (long long)(c1Ok ? c1 : 0) * sXc;

    v8f acc0 = {};
    v8f acc1 = {};
    const int kMain = Cin & ~3;

#pragma unroll 4
    for (int k0 = 0; k0 < kMain; k0 += 4) {
        const int k = k0 + kb;
        const float wa0 = Wr[k];
        const float wa1 = Wr[k + 1];
        const float xb00 = Xc0[(long long)k * sXr];
        const float xb01 = Xc0[(long long)(k + 1) * sXr];
        const float xb10 = Xc1[(long long)k * sXr];
        const float xb11 = Xc1[(long long)(k + 1) * sXr];
        v2f a, b0, b1;
        a.x  = rOk  ? wa0  : 0.0f;
        a.y  = rOk  ? wa1  : 0.0f;
        b0.x = c0Ok ? xb00 : 0.0f;
        b0.y = c0Ok ? xb01 : 0.0f;
        b1.x = c1Ok ? xb10 : 0.0f;
        b1.y = c1Ok ? xb11 : 0.0f;
        acc0 = __builtin_amdgcn_wmma_f32_16x16x4_f32(false, a, false, b0,
                                                     (short)0, acc0, false, false);
        acc1 = __builtin_amdgcn_wmma_f32_16x16x4_f32(false, a, false, b1,
                                                     (short)0, acc1, false, false);
    }

    if (kMain < Cin) {   // uniform remainder chunk (Cin % 4 != 0)
        const int  k    = kMain + kb;
        const bool k0Ok = (k < Cin);
        const bool k1Ok = (k + 1 < Cin);
        const int  kc0  = k0Ok ? k : 0;
        const int  kc1  = k1Ok ? k + 1 : 0;
        const float wa0 = Wr[kc0];
        const float wa1 = Wr[kc1];
        const float xb00 = Xc0[(long long)kc0 * sXr];
        const float xb01 = Xc0[(long long)kc1 * sXr];
        const float xb10 = Xc1[(long long)kc0 * sXr];
        const float xb11 = Xc1[(long long)kc1 * sXr];
        v2f a, b0, b1;
        a.x  = (rOk && k0Ok)  ? wa0  : 0.0f;
        a.y  = (rOk && k1Ok)  ? wa1  : 0.0f;
        b0.x = (c0Ok && k0Ok) ? xb00 : 0.0f;
        b0.y = (c0Ok && k1Ok) ? xb01 : 0.0f;
        b1.x = (c1Ok && k0Ok) ? xb10 : 0.0f;
        b1.y = (c1Ok && k1Ok) ? xb11 : 0.0f;
        acc0 = __builtin_amdgcn_wmma_f32_16x16x4_f32(false, a, false, b0,
                                                     (short)0, acc0, false, false);
        acc1 = __builtin_amdgcn_wmma_f32_16x16x4_f32(false, a, false, b1,
                                                     (short)0, acc1, false, false);
    }

    // ---- epilogue: store both 16-column sub-tiles --------------------------
    const int half = (lane >> 4) * 8;
    float* Yb = Y + (long long)b * sYb;
#pragma unroll
    for (int s = 0; s < 2; ++s) {
        const int cc = (s == 0) ? c0 : c1;
        if (cc < M) {
            const int dd = (Np > 0) ? (cc / Np) : 0;
#pragma unroll
            for (int i = 0; i < 8; ++i) {
                const int rr = row0 + half + i;
                if (rr < Cout) {
                    float v = (s == 0) ? acc0[i] : acc1[i];
                    if (rowBias) v += rowBias[rr];
                    if (posBias) v += posBias[((long long)b * Cout + rr) * 3 + dd];
                    if (relu)    v = fmaxf(v, 0.0f);
                    Yb[(long long)rr * M + cc] = v;
                }
            }
        }
    }
}

// ===========================================================================
// kNN: fused pairwise distance + top-20 (largest neg_dist) per point.
// One block of 256 threads per 256 points; x[b] staged in LDS (24 KB).
// ===========================================================================
__global__ void knn_kernel(const float* __restrict__ x, int* __restrict__ idx)
{
    __shared__ float xs[3 * NPTS];
    const int b   = blockIdx.y;
    const int tid = threadIdx.x;
    for (int t = tid; t < 3 * NPTS; t += blockDim.x)
        xs[t] = x[(long long)b * 3 * NPTS + t];
    __syncthreads();

    const int n = blockIdx.x * blockDim.x + tid;
    const float cx = xs[n], cy = xs[NPTS + n], cz = xs[2 * NPTS + n];

    float best[KNN]; int bidx[KNN];
#pragma unroll
    for (int i = 0; i < KNN; ++i) { best[i] = -3.0e38f; bidx[i] = 0; }

    for (int m = 0; m < NPTS; ++m) {
        const float dx = cx - xs[m];
        const float dy = cy - xs[NPTS + m];
        const float dz = cz - xs[2 * NPTS + m];
        const float d  = -(dx * dx + dy * dy + dz * dz);
        if (d > best[KNN - 1]) {
            int j = KNN - 1;
            while (j > 0 && best[j - 1] < d) {
                best[j] = best[j - 1]; bidx[j] = bidx[j - 1]; --j;
            }
            best[j] = d; bidx[j] = m;
        }
    }
    int* o = idx + ((long long)b * NPTS + n) * KNN;
#pragma unroll
    for (int i = 0; i < KNN; ++i) o[i] = bidx[i];
}

// ===========================================================================
// 'pos' stage pass 1: recompute graph features (edge, center, cross),
// f = Wf(21x3) * feat, accumulate per-channel sum / sumsq of ||f||+eps.
// ===========================================================================
__global__ void pos_stats_kernel(const float* __restrict__ x, const int* __restrict__ idx,
                                 const float* __restrict__ Wf,
                                 float* __restrict__ sum, float* __restrict__ sumsq)
{
    __shared__ float ssum[21], ssq[21];
    const int tid = threadIdx.x;
    if (tid < 21) { ssum[tid] = 0.0f; ssq[tid] = 0.0f; }
    __syncthreads();

    const int b = blockIdx.y;
    const int n = blockIdx.x * blockDim.x + tid;
    const float* xb = x + (long long)b * 3 * NPTS;
    const float cx = xb[n], cy = xb[NPTS + n], cz = xb[2 * NPTS + n];

    float lsum[21], lsq[21];
#pragma unroll
    for (int o = 0; o < 21; ++o) { lsum[o] = 0.0f; lsq[o] = 0.0f; }

    const int* ib = idx + ((long long)b * NPTS + n) * KNN;
    for (int k = 0; k < KNN; ++k) {
        const int m = ib[k];
        const float nx = xb[m], ny = xb[NPTS + m], nz = xb[2 * NPTS + m];
        const float ex = nx - cx, ey = ny - cy, ez = nz - cz;
        const float rx = ny * cz - nz * cy;
        const float ry = nz * cx - nx * cz;
        const float rz = nx * cy - ny * cx;
#pragma unroll
        for (int o = 0; o < 21; ++o) {
            const float w0 = Wf[o * 3], w1 = Wf[o * 3 + 1], w2 = Wf[o * 3 + 2];
            const float fx = w0 * ex + w1 * cx + w2 * rx;
            const float fy = w0 * ey + w1 * cy + w2 * ry;
            const float fz = w0 * ez + w1 * cz + w2 * rz;
            const float nn = sqrtf(fx * fx + fy * fy + fz * fz) + EPSV;
            lsum[o] += nn; lsq[o] += nn * nn;
        }
    }
#pragma unroll
    for (int o = 0; o < 21; ++o) {
        atomicAdd(&ssum[o], lsum[o]);
        atomicAdd(&ssq[o],  lsq[o]);
    }
    __syncthreads();
    if (tid < 21) { atomicAdd(&sum[tid], ssum[tid]); atomicAdd(&sumsq[tid], ssq[tid]); }
}

// ===========================================================================
// 'pos' stage pass 2: recompute features, apply vn_bn + VN projection
// nonlinearity, mean over k; output h0 (B,21,3,N).
// ===========================================================================
__global__ void pos_apply_kernel(const float* __restrict__ x, const int* __restrict__ idx,
                                 const float* __restrict__ Wf, const float* __restrict__ Wd,
                                 const float* __restrict__ mean, const float* __restrict__ invstd,
                                 const float* __restrict__ gamma, const float* __restrict__ beta,
                                 float* __restrict__ out)
{
    const int b = blockIdx.y;
    const int n = blockIdx.x * blockDim.x + threadIdx.x;
    const float* xb = x + (long long)b * 3 * NPTS;
    const float cx = xb[n], cy = xb[NPTS + n], cz = xb[2 * NPTS + n];

    float acc[63];
#pragma unroll
    for (int i = 0; i < 63; ++i) acc[i] = 0.0f;

    const int* ib = idx + ((long long)b * NPTS + n) * KNN;
    for (int k = 0; k < KNN; ++k) {
        const int m = ib[k];
        const float nx = xb[m], ny = xb[NPTS + m], nz = xb[2 * NPTS + m];
        const float ex = nx - cx, ey = ny - cy, ez = nz - cz;
        const float rx = ny * cz - nz * cy;
        const float ry = nz * cx - nx * cz;
        const float rz = nx * cy - ny * cx;
#pragma unroll
        for (int o = 0; o < 21; ++o) {
            const float f0 = Wf[o * 3], f1 = Wf[o * 3 + 1], f2 = Wf[o * 3 + 2];
            float fx = f0 * ex + f1 * cx + f2 * rx;
            float fy = f0 * ey + f1 * cy + f2 * ry;
            float fz = f0 * ez + f1 * cz + f2 * rz;
            const float g0 = Wd[o * 3], g1 = Wd[o * 3 + 1], g2 = Wd[o * 3 + 2];
            const float dx = g0 * ex + g1 * cx + g2 * rx;
            const float dy = g0 * ey + g1 * cy + g2 * ry;
            const float dz = g0 * ez + g1 * cz + g2 * rz;

            const float nn  = sqrtf(fx * fx + fy * fy + fz * fz) + EPSV;
            const float nbn = (nn - mean[o]) * invstd[o] * gamma[o] + beta[o];
            const float s   = nbn / nn;
            fx *= s; fy *= s; fz *= s;
            const float dot = fx * dx + fy * dy + fz * dz;
            if (dot < 0.0f) {
                const float t = dot / (dx * dx + dy * dy + dz * dz + EPSV);
                fx -= t * dx; fy -= t * dy; fz -= t * dz;
            }
            acc[o * 3 + 0] += fx; acc[o * 3 + 1] += fy; acc[o * 3 + 2] += fz;
        }
    }
    const float inv = 1.0f / (float)KNN;
#pragma unroll
    for (int o = 0; o < 21; ++o) {
        const long long base = (((long long)b * 21 + o) * 3) * NPTS + n;
        out[base]             = acc[o * 3 + 0] * inv;
        out[base + NPTS]      = acc[o * 3 + 1] * inv;
        out[base + 2 * NPTS]  = acc[o * 3 + 2] * inv;
    }
}

// ===========================================================================
// vn_bn statistics over (b, n) per channel:  norm = ||F[b,c,:,n]|| + eps
// ===========================================================================
__global__ void bn_stats_kernel(const float* __restrict__ F,
                                float* __restrict__ sum, float* __restrict__ sumsq,
                                int C, int Np)
{
    __shared__ float r1[256], r2[256];
    const int c   = blockIdx.y;
    const int b   = blockIdx.z;
    const int tid = threadIdx.x;
    const int n   = blockIdx.x * 256 + tid;
    float nn = 0.0f, n2 = 0.0f;
    if (n < Np) {
        const long long base = (((long long)b * C + c) * 3) * Np + n;
        const float x0 = F[base], x1 = F[base + Np], x2 = F[base + 2 * Np];
        nn = sqrtf(x0 * x0 + x1 * x1 + x2 * x2) + EPSV;
        n2 = nn * nn;
    }
    r1[tid] = nn; r2[tid] = n2;
    __syncthreads();
    for (int s = 128; s > 0; s >>= 1) {
        if (tid < s) { r1[tid] += r1[tid + s]; r2[tid] += r2[tid + s]; }
        __syncthreads();
    }
    if (tid == 0) { atomicAdd(&sum[c], r1[0]); atomicAdd(&sumsq[c], r2[0]); }
}

__global__ void bn_finalize_kernel(const float* __restrict__ sum, const float* __restrict__ sumsq,
                                   float cnt, float* __restrict__ mean, float* __restrict__ invstd,
                                   int C)
{
    const int c = blockIdx.x * blockDim.x + threadIdx.x;
    if (c < C) {
        const float m = sum[c] / cnt;
        const float v = sumsq[c] / cnt - m * m;
        mean[c]   = m;
        invstd[c] = rsqrtf(v + BN_EPSV);
    }
}

// vn_bn apply (+ optional VN projection nonlinearity), in-place on F.
__global__ void bn_apply_kernel(float* __restrict__ F, const float* __restrict__ D,
                                const float* __restrict__ mean, const float* __restrict__ invstd,
                                const float* __restrict__ gamma, const float* __restrict__ beta,
                                int C, int Np, int nonlin, long long total)
{
    const long long t = (long long)blockIdx.x * blockDim.x + threadIdx.x;
    if (t >= total) return;
    const int n = (int)(t % Np);
    const long long r = t / Np;
    const int c = (int)(r % C);
    const int b = (int)(r / C);
    const long long base = (((long long)b * C + c) * 3) * Np + n;

    float f0 = F[base], f1 = F[base + Np], f2 = F[base + 2 * Np];
    const float nn  = sqrtf(f0 * f0 + f1 * f1 + f2 * f2) + EPSV;
    const float nbn = (nn - mean[c]) * invstd[c] * gamma[c] + beta[c];
    const float s   = nbn / nn;
    f0 *= s; f1 *= s; f2 *= s;
    if (nonlin) {
        const float d0 = D[base], d1 = D[base + Np], d2 = D[base + 2 * Np];
        const float dot = f0 * d0 + f1 * d1 + f2 * d2;
        if (dot < 0.0f) {
            const float tt = dot / (d0 * d0 + d1 * d1 + d2 * d2 + EPSV);
            f0 -= tt * d0; f1 -= tt * d1; f2 -= tt * d2;
        }
    }
    F[base] = f0; F[base + Np] = f1; F[base + 2 * Np] = f2;
}

// mean over N: (B,C,3,Np) -> (B,C,3) laid out b*C*3 + c*3 + d
__global__ void meanN_kernel(const float* __restrict__ F, float* __restrict__ out, int C, int Np)
{
    const int t = blockIdx.x * blockDim.x + threadIdx.x;
    if (t >= BATCH * C * 3) return;
    const int d = t % 3;
    const int rr = t / 3;
    const int c = rr % C;
    const int b = rr / C;
    const long long base = ((((long long)b * C + c) * 3) + d) * Np;
    float s = 0.0f;
    for (int n = 0; n < Np; ++n) s += F[base + n];
    out[t] = s / (float)Np;
}

// x_std = einsum('bijm,bjkm->bikm', concat(h, hmean), z0); max over m -> latent
__global__ void xstd_max_kernel(const float* __restrict__ H, const float* __restrict__ hmean,
                                const float* __restrict__ z0,
                                float* __restrict__ latent, float* __restrict__ latT)
{
    const int t = blockIdx.x * blockDim.x + threadIdx.x;
    if (t >= BATCH * 2046) return;
    const int c = t % 2046;
    const int b = t / 2046;
    const int i  = c / 3;
    const int kk = c % 3;
    const float* zb = z0 + (long long)b * 3 * MCOLS;
    float m = -3.0e38f;
    if (i < 341) {
        const float* hb = H + ((long long)b * 341 + i) * MCOLS;
        for (int n = 0; n < NPTS; ++n) {
            const float s = hb[n]            * zb[(0 * 3 + kk) * NPTS + n]
                          + hb[NPTS + n]     * zb[(1 * 3 + kk) * NPTS + n]
                          + hb[2 * NPTS + n] * zb[(2 * 3 + kk) * NPTS + n];
            m = fmaxf(m, s);
        }
    } else {
        const int im = i - 341;
        const float h0 = hmean[((long long)b * 341 + im) * 3 + 0];
        const float h1 = hmean[((long long)b * 341 + im) * 3 + 1];
        const float h2 = hmean[((long long)b * 341 + im) * 3 + 2];
        for (int n = 0; n < NPTS; ++n) {
            const float s = h0 * zb[(0 * 3 + kk) * NPTS + n]
                          + h1 * zb[(1 * 3 + kk) * NPTS + n]
                          + h2 * zb[(2 * 3 + kk) * NPTS + n];
            m = fmaxf(m, s);
        }
    }
    latent[(long long)b * 2046 + c] = m;
    latT[(long long)c * 16 + b]     = m;
}

// reconstructed[b, q] = Yd2[b, q] + pc1_xyz[b, q/96, q%3]
__global__ void assemble_kernel(const float* __restrict__ Yd2, const float* __restrict__ Yfcx,
                                float* __restrict__ rec)
{
    const int t = blockIdx.x * blockDim.x + threadIdx.x;
    if (t >= BATCH * 6144) return;
    const int q = t % 6144;
    const int b = t / 6144;
    const int cc = q / 96;
    const int dd = q % 3;
    rec[t] = Yd2[t] + Yfcx[(cc * 3 + dd) * 16 + b];
}

__global__ void zero_kernel(float* __restrict__ p, int n)
{
    const int t = blockIdx.x * blockDim.x + threadIdx.x;
    if (t < n) p[t] = 0.0f;
}

// ===========================================================================
// Host orchestration
// ===========================================================================
extern "C" void kernel_launch(void* const* d_in, const int* in_sizes, int n_in,
                              void* d_out, int out_size, void* d_ws, size_t ws_size,
                              hipStream_t stream)
{
    (void)in_sizes; (void)n_in; (void)out_size; (void)ws_size;

    // ---- inputs (setup_inputs dict order: x, then params in insertion order)
    const float* X       = (const float*)d_in[0];
    const float* pos_Wf  = (const float*)d_in[1];
    const float* pos_Wd  = (const float*)d_in[2];
    const float* pos_g   = (const float*)d_in[3];
    const float* pos_b   = (const float*)d_in[4];
    const float* c1_Wf   = (const float*)d_in[5];
    const float* c1_Wd   = (const float*)d_in[6];
    const float* c1_g    = (const float*)d_in[7];
    const float* c1_b    = (const float*)d_in[8];
    const float* c2_Wf   = (const float*)d_in[9];
    const float* c2_Wd   = (const float*)d_in[10];
    const float* c2_g    = (const float*)d_in[11];
    const float* c2_b    = (const float*)d_in[12];
    const float* c3_W    = (const float*)d_in[13];
    const float* bn3_g   = (const float*)d_in[14];
    const float* bn3_b   = (const float*)d_in[15];
    const float* std1_Wf = (const float*)d_in[16];
    const float* std1_Wd = (const float*)d_in[17];
    const float* std1_g  = (const float*)d_in[18];
    const float* std1_b  = (const float*)d_in[19];
    const float* std2_Wf = (const float*)d_in[20];
    const float* std2_Wd = (const float*)d_in[21];
    const float* std2_g  = (const float*)d_in[22];
    const float* std2_b  = (const float*)d_in[23];
    const float* std_lin = (const float*)d_in[24];
    const float* s1_Wf   = (const float*)d_in[25];
    const float* s1_Wd   = (const float*)d_in[26];
    const float* s1_g    = (const float*)d_in[27];
    const float* s1_b    = (const float*)d_in[28];
    const float* s2_Wf   = (const float*)d_in[29];
    const float* s2_Wd   = (const float*)d_in[30];
    const float* s2_g    = (const float*)d_in[31];
    const float* s2_b    = (const float*)d_in[32];
    const float* s3_Wf   = (const float*)d_in[33];
    const float* s3_Wd   = (const float*)d_in[34];
    const float* s3_g    = (const float*)d_in[35];
    const float* s3_b    = (const float*)d_in[36];
    const float* s4_Wf   = (const float*)d_in[37];
    const float* s4_Wd   = (const float*)d_in[38];
    const float* s4_g    = (const float*)d_in[39];
    const float* s4_b    = (const float*)d_in[40];
    const float* s5_Wf   = (const float*)d_in[41];
    const float* s5_Wd   = (const float*)d_in[42];
    const float* s5_g    = (const float*)d_in[43];
    const float* s5_b    = (const float*)d_in[44];
    const float* s6_W    = (const float*)d_in[45];
    const float* fc1_W   = (const float*)d_in[46];
    const float* fc1_b   = (const float*)d_in[47];
    const float* fcx_W   = (const float*)d_in[48];
    const float* fcx_b   = (const float*)d_in[49];
    const float* d1_W    = (const float*)d_in[50];
    const float* d1_b    = (const float*)d_in[51];
    const float* d2_W    = (const float*)d_in[52];
    const float* d2_b    = (const float*)d_in[53];

    // ---- output regions
    float* OUT_REC = (float*)d_out;                 // (16,3,2048)   = 98304
    float* OUT_LAT = OUT_REC + 98304;               // (16,2046)     = 32736
    float* OUT_Z0  = OUT_LAT + 32736;               // (16,3,3,2048) = 294912

    // ---- workspace layout (floats)
    const size_t SZ21  = (size_t)BATCH * 21  * MCOLS;   // 2,064,384
    const size_t SZ42  = (size_t)BATCH * 42  * MCOLS;   // 4,128,768
    const size_t SZ341 = (size_t)BATCH * 341 * MCOLS;   // 33,521,664
    const size_t SZ170 = (size_t)BATCH * 170 * MCOLS;   // 16,711,680

    float* wsf = (float*)d_ws;
    int*   IDX = (int*)d_ws;                            // 655,360 ints
    size_t off = 655360;
    float* B21A = wsf + off; off += SZ21;
    float* B21B = wsf + off; off += SZ21;
    float* B21C = wsf + off; off += SZ21;
    float* B42A = wsf + off; off += SZ42;
    float* B42B = wsf + off; off += SZ42;
    float* G1   = wsf + off; off += SZ341;
    float* G2   = wsf + off; off += SZ341;
    float* G3   = wsf + off; off += SZ341;
    float* M1   = wsf + off; off += SZ170;
    float* M2   = wsf + off; off += SZ170;
    float* SUM  = wsf + off; off += 704;
    float* SSQ  = wsf + off; off += 704;
    float* MEAN = wsf + off; off += 704;
    float* ISTD = wsf + off; off += 704;
    float* SM    = wsf + off; off += 16368;  // stn mean  (16,341,3)
    float* HMEAN = wsf + off; off += 16368;  // h mean    (16,341,3)
    float* BIASF = wsf + off; off += 16368;  // concat-fold bias (F path)
    float* BIASD = wsf + off; off += 16368;  // concat-fold bias (D path)
    float* S4F = wsf + off; off += 8160;
    float* S4D = wsf + off; off += 8160;
    float* S5F = wsf + off; off += 4080;
    float* S5D = wsf + off; off += 4080;
    float* XG  = wsf + off; off += 1008;     // (16,21,3)
    float* LATT = wsf + off; off += 32736;   // latent^T (2046,16)
    float* YFC1 = wsf + off; off += 262144;  // (16384,16)
    float* YFCX = wsf + off; off += 3072;    // (192,16)
    float* YD1  = wsf + off; off += 262144;  // (16,256,64)
    float* YD2  = wsf + off; off += 98304;   // (16,96,64)

    // ---- helpers -----------------------------------------------------------
    auto gemm = [&](const float* W, int ldW, const float* Xp,
                    long long sXb, long long sXr, long long sXc,
                    float* Y, long long sYb, const float* rowB, const float* posB,
                    int Cout, int Cin, int M, int Np, int relu, int batch) {
        dim3 g((M + 127) / 128, (Cout + 15) / 16, batch);   // 4 waves x 32 cols
        gemm_wmma_kernel<<<g, dim3(128), 0, stream>>>(W, ldW, Xp, sXb, sXr, sXc,
                                                      Y, sYb, rowB, posB,
                                                      Cout, Cin, M, Np, relu);
    };
    auto bnstage = [&](float* F, const float* D, const float* g, const float* bt,
                       int C, int Np, int nonlin) {
        zero_kernel<<<(1408 + 255) / 256, 256, 0, stream>>>(SUM, 1408); // SUM+SSQ
        dim3 gs((Np + 255) / 256, C, BATCH);
        bn_stats_kernel<<<gs, 256, 0, stream>>>(F, SUM, SSQ, C, Np);
        bn_finalize_kernel<<<(C + 63) / 64, 64, 0, stream>>>(SUM, SSQ,
            (float)((long long)BATCH * Np), MEAN, ISTD, C);
        const long long tot = (long long)BATCH * C * Np;
        bn_apply_kernel<<<(unsigned)((tot + 255) / 256), 256, 0, stream>>>(
            F, D, MEAN, ISTD, g, bt, C, Np, nonlin, tot);
    };
    const long long SR = 3LL * NPTS; // channel stride for (B,C,3,N) tensors

    // ---- 1. kNN graph ------------------------------------------------------
    knn_kernel<<<dim3(NPTS / 256, BATCH), 256, 0, stream>>>(X, IDX);

    // ---- 2. 'pos' block (fused graph features, BN, nonlin, mean over k) ----
    zero_kernel<<<(1408 + 255) / 256, 256, 0, stream>>>(SUM, 1408);
    pos_stats_kernel<<<dim3(NPTS / 256, BATCH), 256, 0, stream>>>(X, IDX, pos_Wf, SUM, SSQ);
    bn_finalize_kernel<<<1, 64, 0, stream>>>(SUM, SSQ,
        (float)((long long)BATCH * NPTS * KNN), MEAN, ISTD, 21);
    pos_apply_kernel<<<dim3(NPTS / 256, BATCH), 256, 0, stream>>>(
        X, IDX, pos_Wf, pos_Wd, MEAN, ISTD, pos_g, pos_b, B21A);   // h0 -> B21A

    // ---- 3. c1 block (21 -> 21) -------------------------------------------
    gemm(c1_Wf, 21, B21A, 21 * SR, SR, 1, B21B, 21 * SR, nullptr, nullptr, 21, 21, MCOLS, 0, 0, BATCH);
    gemm(c1_Wd, 21, B21A, 21 * SR, SR, 1, B21C, 21 * SR, nullptr, nullptr, 21, 21, MCOLS, 0, 0, BATCH);
    bnstage(B21B, B21C, c1_g, c1_b, 21, NPTS, 1);                  // h1 = B21B

    // ---- 4. STN ------------------------------------------------------------
    gemm(s1_Wf, 21, B21B, 21 * SR, SR, 1, B21A, 21 * SR, nullptr, nullptr, 21, 21, MCOLS, 0, 0, BATCH);
    gemm(s1_Wd, 21, B21B, 21 * SR, SR, 1, B21C, 21 * SR, nullptr, nullptr, 21, 21, MCOLS, 0, 0, BATCH);
    bnstage(B21A, B21C, s1_g, s1_b, 21, NPTS, 1);                  // t1 = B21A

    gemm(s2_Wf, 21, B21A, 21 * SR, SR, 1, B42A, 42 * SR, nullptr, nullptr, 42, 21, MCOLS, 0, 0, BATCH);
    gemm(s2_Wd, 21, B21A, 21 * SR, SR, 1, B42B, 42 * SR, nullptr, nullptr, 42, 21, MCOLS, 0, 0, BATCH);
    bnstage(B42A, B42B, s2_g, s2_b, 42, NPTS, 1);                  // t2 = B42A

    gemm(s3_Wf, 42, B42A, 42 * SR, SR, 1, G1, 341 * SR, nullptr, nullptr, 341, 42, MCOLS, 0, 0, BATCH);
    gemm(s3_Wd, 42, B42A, 42 * SR, SR, 1, G2, 341 * SR, nullptr, nullptr, 341, 42, MCOLS, 0, 0, BATCH);
    bnstage(G1, G2, s3_g, s3_b, 341, NPTS, 1);                     // t3 = G1

    meanN_kernel<<<(BATCH * 341 * 3 + 255) / 256, 256, 0, stream>>>(G1, SM, 341, NPTS);

    gemm(s4_Wf, 341, SM, 1023, 3, 1, S4F, 510, nullptr, nullptr, 170, 341, 3, 0, 0, BATCH);
    gemm(s4_Wd, 341, SM, 1023, 3, 1, S4D, 510, nullptr, nullptr, 170, 341, 3, 0, 0, BATCH);
    bnstage(S4F, S4D, s4_g, s4_b, 170, 1, 1);

    gemm(s5_Wf, 170, S4F, 510, 3, 1, S5F, 255, nullptr, nullptr, 85, 170, 3, 0, 0, BATCH);
    gemm(s5_Wd, 170, S4F, 510, 3, 1, S5D, 255, nullptr, nullptr, 85, 170, 3, 0, 0, BATCH);
    bnstage(S5F, S5D, s5_g, s5_b, 85, 1, 1);

    gemm(s6_W, 85, S5F, 255, 3, 1, XG, 63, nullptr, nullptr, 21, 85, 3, 0, 0, BATCH); // xg

    // ---- 5. c2 block on concat([h1, xg]) : fold xg half into a pos-bias ----
    gemm(c2_Wf + 21, 42, XG, 63, 3, 1, BIASF, 126, nullptr, nullptr, 42, 21, 3, 0, 0, BATCH);
    gemm(c2_Wd + 21, 42, XG, 63, 3, 1, BIASD, 126, nullptr, nullptr, 42, 21, 3, 0, 0, BATCH);
    gemm(c2_Wf, 42, B21B, 21 * SR, SR, 1, B42A, 42 * SR, nullptr, BIASF, 42, 21, MCOLS, NPTS, 0, BATCH);
    gemm(c2_Wd, 42, B21B, 21 * SR, SR, 1, B42B, 42 * SR, nullptr, BIASD, 42, 21, MCOLS, NPTS, 0, BATCH);
    bnstage(B42A, B42B, c2_g, c2_b, 42, NPTS, 1);                  // h2 = B42A

    // ---- 6. c3 + bn3 (no nonlinearity) ------------------------------------
    gemm(c3_W, 42, B42A, 42 * SR, SR, 1, G1, 341 * SR, nullptr, nullptr, 341, 42, MCOLS, 0, 0, BATCH);
    bnstage(G1, nullptr, bn3_g, bn3_b, 341, NPTS, 0);              // h_c3 = G1
    meanN_kernel<<<(BATCH * 341 * 3 + 255) / 256, 256, 0, stream>>>(G1, HMEAN, 341, NPTS);

    // ---- 7. std1 block on concat([h_c3, hmean]) ----------------------------
    gemm(std1_Wf + 341, 682, HMEAN, 1023, 3, 1, BIASF, 1023, nullptr, nullptr, 341, 341, 3, 0, 0, BATCH);
    gemm(std1_Wd + 341, 682, HMEAN, 1023, 3, 1, BIASD, 1023, nullptr, nullptr, 341, 341, 3, 0, 0, BATCH);
    gemm(std1_Wf, 682, G1, 341 * SR, SR, 1, G2, 341 * SR, nullptr, BIASF, 341, 341, MCOLS, NPTS, 0, BATCH);
    gemm(std1_Wd, 682, G1, 341 * SR, SR, 1, G3, 341 * SR, nullptr, BIASD, 341, 341, MCOLS, NPTS, 0, BATCH);
    bnstage(G2, G3, std1_g, std1_b, 341, NPTS, 1);                 // z1 = G2

    // ---- 8. std2 block (341 -> 170) ---------------------------------------
    gemm(std2_Wf, 341, G2, 341 * SR, SR, 1, M1, 170 * SR, nullptr, nullptr, 170, 341, MCOLS, 0, 0, BATCH);
    gemm(std2_Wd, 341, G2, 341 * SR, SR, 1, M2, 170 * SR, nullptr, nullptr, 170, 341, MCOLS, 0, 0, BATCH);
    bnstage(M1, M2, std2_g, std2_b, 170, NPTS, 1);                 // z = M1

    // ---- 9. z0 = std_lin * z  -> output region 3 ---------------------------
    gemm(std_lin, 170, M1, 170 * SR, SR, 1, OUT_Z0, 3 * SR, nullptr, nullptr, 3, 170, MCOLS, 0, 0, BATCH);

    // ---- 10. x_std einsum + max over N -> latent ---------------------------
    xstd_max_kernel<<<(BATCH * 2046 + 255) / 256, 256, 0, stream>>>(G1, HMEAN, OUT_Z0, OUT_LAT, LATT);

    // ---- 11. decoder -------------------------------------------------------
    gemm(fc1_W, 2046, LATT, 0, 16, 1, YFC1, 0, fc1_b, nullptr, 16384, 2046, 16, 0, 0, 1);
    gemm(fcx_W, 2046, LATT, 0, 16, 1, YFCX, 0, fcx_b, nullptr, 192,   2046, 16, 0, 0, 1);
    // d1: X[b][c][j] = YFC1[(j*256+c)*16 + b]  -> strides (b:1, c:16, j:4096), relu
    gemm(d1_W, 256, YFC1, 1, 16, 4096, YD1, 16384, d1_b, nullptr, 256, 256, 64, 0, 1, BATCH);
    // d2: X = YD1 (B,256,64) row-major
    gemm(d2_W, 256, YD1, 16384, 64, 1, YD2, 6144, d2_b, nullptr, 96, 256, 64, 0, 0, BATCH);

    assemble_kernel<<<(BATCH * 6144 + 255) / 256, 256, 0, stream>>>(YD2, YFCX, OUT_REC);
}